// AlmostFairKCRPSLoss_73538430042412
// MI455X (gfx1250) — compile-verified
//
#include <hip/hip_runtime.h>

// CDNA5 / gfx1250, wave32.
// AlmostFairKCRPSLoss with ALPHA=1.0 (epsilon=0):
//   crps(s) = (1/m)*sum_i |p_i - t|  -  (1/(m*(m-1))) * sum_{i<j} |p_i - p_j|
//   out = mean_s crps(s)
// Roofline: ~45 MB read -> 1.9 us HBM floor; working set < 192 MB L2 so
// replays are L2-resident and VALU-bound. Pairwise L1 uses the order-statistic
// identity  sum_{i<j}|p_i-p_j| = sum_k (2k-m+1)*p_(k)  with a Batcher
// odd-even mergesort network (63 min/max pairs for m=16): ~174 VALU/location
// vs ~272 for the O(m^2) form. V_WMMA_F32_16X16X4_F32 performs the exact
// f32 cross-lane sum reduction; ds_swizzle XOR folds finish the wave.

typedef __attribute__((ext_vector_type(2))) float v2f;
typedef __attribute__((ext_vector_type(8))) float v8f;

// One ds_swizzle_b32 XOR-fold step (EXEC must be full at call site).
// offset encoding (group-of-32): [14:10]=xor_mask, [9:5]=or=0, [4:0]=and=0x1f.
template <int XOR_MASK>
__device__ __forceinline__ float fold_xor_add(float x) {
  union { float f; int i; } u, r;
  u.f = x;
  r.i = __builtin_amdgcn_ds_swizzle(u.i, (XOR_MASK << 10) | 0x1f);
  return x + r.f;
}

// Exact 32-lane f32 sum via V_WMMA_F32_16X16X4_F32.
// A = ones(16x4); each lane's value occupies exactly one B slot (other zeroed):
//   D[m][n] = sum_k B[k][n]  (independent of m; robust to exact B striping).
// C/D layout: VGPR0 lanes 0-15 = D[0][n], lanes 16-31 = D[8][n]. Four XOR
// swizzle folds within each 16-lane half sum the 16 column sums -> the full
// 32-lane total in every lane. Requires EXEC all ones.
__device__ __forceinline__ float wave_reduce_sum_wmma(float x) {
  v2f a; a.x = 1.0f; a.y = 1.0f;
  v2f b; b.x = x;    b.y = 0.0f;
  v8f c = {};
  v8f d = __builtin_amdgcn_wmma_f32_16x16x4_f32(
      /*neg_a=*/false, a, /*neg_b=*/false, b,
      /*c_mod=*/(short)0, c, /*reuse_a=*/false, /*reuse_b=*/false);
  float s = d.s0;
  s = fold_xor_add<1>(s);
  s = fold_xor_add<2>(s);
  s = fold_xor_add<4>(s);
  s = fold_xor_add<8>(s);
  return s;
}

// Deterministic block reduction: WMMA wave sums -> LDS -> thread 0 adds the
// (<=8) wave sums in fixed order and stores the (scaled) block partial.
__device__ __forceinline__ void block_reduce_store(float v,
                                                   float* __restrict__ dst,
                                                   float scale) {
  float wsum = wave_reduce_sum_wmma(v);
  __shared__ float lds[32];
  const int wid = threadIdx.x >> 5;
  const int lid = threadIdx.x & 31;
  if (lid == 0) lds[wid] = wsum;
  __syncthreads();
  if (threadIdx.x == 0) {
    float b = 0.0f;
    const int nw = (blockDim.x + 31) >> 5;
    for (int w = 0; w < nw; ++w) b += lds[w];
    *dst = b * scale;
  }
}

// Sum of |p_i - p_j| over unordered pairs via sorting network + weighted sum.
// Batcher odd-even mergesort (M must be a power of two); sorts p ascending.
// All loop bounds / guards constant-fold at compile time -> 63 CEs for M=16.
template <int M>
__device__ __forceinline__ float pairwise_l1_sorted(float (&p)[M]) {
#pragma unroll
  for (int ph = 1; ph < M; ph <<= 1) {
#pragma unroll
    for (int k = ph; k >= 1; k >>= 1) {
#pragma unroll
      for (int j = k % ph; j <= M - 1 - k; j += 2 * k) {
        const int imax = (k - 1 < M - j - k - 1) ? (k - 1) : (M - j - k - 1);
#pragma unroll
        for (int i = 0; i <= imax; ++i) {
          if ((i + j) / (2 * ph) == (i + j + k) / (2 * ph)) {
            const float lo = fminf(p[i + j], p[i + j + k]);
            const float hi = fmaxf(p[i + j], p[i + j + k]);
            p[i + j] = lo;
            p[i + j + k] = hi;
          }
        }
      }
    }
  }
  // sum_{i<j} (p_(j) - p_(i)) = sum_k (2k - (M-1)) * p_(k)
  float pairs = 0.0f;
#pragma unroll
  for (int k = 0; k < M; ++k)
    pairs = fmaf((float)(2 * k - (M - 1)), p[k], pairs);
  return pairs;
}

// Per-location CRPS from M register-resident values (p is clobbered/sorted).
template <int M>
__device__ __forceinline__ float crps_loc(float (&p)[M], float t) {
  float skill = 0.0f;
#pragma unroll
  for (int i = 0; i < M; ++i) skill += fabsf(p[i] - t);
  const float pairs = pairwise_l1_sorted<M>(p);
  // spread = 2*pairs/(2*M*(M-1)); (1-eps) = 1 for ALPHA=1
  return skill * (1.0f / (float)M) - pairs * (1.0f / (float)(M * (M - 1)));
}

// Vectorized main kernel: each thread covers 4 consecutive locations with
// float4 (global_load_b128) loads. Requires S % 4 == 0 (663552 qualifies).
template <int M>
__global__ void crps_partial_v4_kernel(const float* __restrict__ target,
                                       const float* __restrict__ pred,
                                       float* __restrict__ partials, int S4,
                                       int S) {
  const int g = blockIdx.x * blockDim.x + threadIdx.x;  // float4-group index
  float v = 0.0f;
  if (g < S4) {
    const float4 t4 = ((const float4*)target)[g];
    float4 p4[M];
#pragma unroll
    for (int i = 0; i < M; ++i)
      p4[i] = *(const float4*)(pred + (size_t)i * (size_t)S + 4u * (size_t)g);

#pragma unroll
    for (int c = 0; c < 4; ++c) {
      const float t = (&t4.x)[c];
      float p[M];
#pragma unroll
      for (int i = 0; i < M; ++i) p[i] = (&p4[i].x)[c];
      v += crps_loc<M>(p, t);
    }
  }
  block_reduce_store(v, &partials[blockIdx.x], 1.0f);
}

// Scalar fallback for unexpected shapes (2 <= m <= 32, any S).
__global__ void crps_partial_dyn_kernel(const float* __restrict__ target,
                                        const float* __restrict__ pred,
                                        float* __restrict__ partials, int S,
                                        int M, float invM, float invMM1) {
  const int s = blockIdx.x * blockDim.x + threadIdx.x;
  float v = 0.0f;
  if (s < S && M >= 2 && M <= 32) {
    const float t = target[s];
    float p[32];
    for (int i = 0; i < M; ++i) p[i] = pred[(size_t)i * (size_t)S + (size_t)s];
    float skill = 0.0f;
    for (int i = 0; i < M; ++i) skill += fabsf(p[i] - t);
    float pairs = 0.0f;
    for (int i = 0; i < M; ++i)
      for (int j = i + 1; j < M; ++j) pairs += fabsf(p[i] - p[j]);
    v = skill * invM - pairs * invMM1;
  }
  block_reduce_store(v, &partials[blockIdx.x], 1.0f);
}

// Single-block finish: fixed-order strided accumulation of block partials,
// WMMA wave reduce, LDS fold, global mean scale. Fully deterministic.
__global__ void crps_final_kernel(const float* __restrict__ partials, int n,
                                  float inv_total, float* __restrict__ out) {
  float acc = 0.0f;
  for (int i = threadIdx.x; i < n; i += blockDim.x) acc += partials[i];
  // All threads reconverged here; EXEC full for the WMMA reduction.
  block_reduce_store(acc, out, inv_total);
}

extern "C" void kernel_launch(void* const* d_in, const int* in_sizes, int n_in,
                              void* d_out, int out_size, void* d_ws,
                              size_t ws_size, hipStream_t stream) {
  const float* target = (const float*)d_in[0];
  const float* pred = (const float*)d_in[1];
  float* out = (float*)d_out;

  const int S = in_sizes[0];      // 663552 = 6*2*192*288 (B=1)
  const int m = in_sizes[1] / S;  // 16

  float* partials = (float*)d_ws;
  const int block = 256;  // 8 wave32 waves

  int grid;
  if (m == 16 && (S & 3) == 0) {
    const int S4 = S >> 2;                // 165888
    grid = (S4 + block - 1) / block;      // 648 (exact)
    crps_partial_v4_kernel<16>
        <<<grid, block, 0, stream>>>(target, pred, partials, S4, S);
  } else {
    grid = (S + block - 1) / block;
    crps_partial_dyn_kernel<<<grid, block, 0, stream>>>(
        target, pred, partials, S, m, 1.0f / (float)m,
        1.0f / (float)(m * (m - 1)));
  }

  crps_final_kernel<<<1, 256, 0, stream>>>(partials, grid, 1.0f / (float)S,
                                           out);
}